// DeepSpeedSparseSelfAttention_3307124818262
// MI455X (gfx1250) — compile-verified
//
#include <hip/hip_runtime.h>

typedef __attribute__((ext_vector_type(2))) float v2f;
typedef __attribute__((ext_vector_type(8))) float v8f;

#define SEQ_L 2048
#define HEAD_D 64
#define NBLK 128   // SEQ_L / 16

// One wave (32 lanes) computes one 16-row query block for one (b,h),
// iterating its statically-known sparse block columns with flash-attention
// online softmax. All matmuls via v_wmma_f32_16x16x4_f32 (fp32 WMMA).
__global__ __launch_bounds__(32)
void sparse_attn_wmma_f32(const float* __restrict__ q,
                          const float* __restrict__ k,
                          const float* __restrict__ v,
                          float* __restrict__ out) {
  __shared__ float pshm[16 * 16];  // P tile bounce buffer (C/D layout -> A layout)

  const int lane = threadIdx.x;        // 0..31
  const int half = lane >> 4;          // 0 or 1
  const int l16  = lane & 15;

  const int  row  = (int)(blockIdx.x & (NBLK - 1));   // query block row
  const long bh   = (long)(blockIdx.x >> 7);          // b*H + h
  const long base = bh * (long)(SEQ_L * HEAD_D);

  // ---- preload Q block in WMMA-A layout, pre-scaled by 1/sqrt(D) ----
  // A 16x4 f32: lane holds row m=l16; VGPR r holds K = r + 2*half.
  const float* qrow = q + base + (long)(row * 16 + l16) * HEAD_D + 2 * half;
  v2f qa[16];
#pragma unroll
  for (int kk = 0; kk < 16; ++kk) {
    v2f t = *(const v2f*)(qrow + 4 * kk);
    qa[kk].x = t.x * 0.125f;
    qa[kk].y = t.y * 0.125f;
  }

  // O accumulator: 4 chunks of 16 columns, each in C/D layout (row = r + 8*half)
  v8f o0 = {}, o1 = {}, o2 = {}, o3 = {};
  float mrow[8], lrow[8];
#pragma unroll
  for (int r = 0; r < 8; ++r) { mrow[r] = -1e30f; lrow[r] = 0.0f; }

  // static fixed layout: verticals j=3,7,..,<w then locals w..row  (w = row&~3)
  const int w     = row & ~3;
  const int nvert = w >> 2;
  const int nbl   = nvert + (row - w + 1);

  for (int t = 0; t < nbl; ++t) {
    const int jb = (t < nvert) ? (4 * t + 3) : (w + (t - nvert));

    // speculative prefetch of next block's K/V tiles
    {
      const int tn  = (t + 1 < nbl) ? (t + 1) : t;
      const int jbn = (tn < nvert) ? (4 * tn + 3) : (w + (tn - nvert));
      const float* kp = k + base + (long)(jbn * 16 + l16) * HEAD_D;
      const float* vp = v + base + (long)(jbn * 16 + l16) * HEAD_D;
      __builtin_prefetch(kp, 0, 0);
      __builtin_prefetch(vp, 0, 0);
    }

    // ---- S = (Q * 1/sqrt(D)) * K^T  (16 steps of K=4) ----
    // B 4x16 f32: lane holds col n=l16 (key index); VGPR r = K-row r + 2*half.
    // B[j][n] = K[n][4kk+j]  -> contiguous float2 per lane.
    const float* krow = k + base + (long)(jb * 16 + l16) * HEAD_D + 2 * half;
    v8f s = {};
#pragma unroll
    for (int kk = 0; kk < 16; ++kk) {
      v2f b = *(const v2f*)(krow + 4 * kk);
      s = __builtin_amdgcn_wmma_f32_16x16x4_f32(false, qa[kk], false, b,
                                                (short)0, s, false, false);
    }

    // ---- online softmax: row r of this lane-half is m = r + 8*half ----
    float alpha[8];
#pragma unroll
    for (int r = 0; r < 8; ++r) {
      float x  = s[r];
      float rm = x;
#pragma unroll
      for (int off = 1; off < 16; off <<= 1)
        rm = fmaxf(rm, __shfl_xor(rm, off, 32));
      const float mnew = fmaxf(mrow[r], rm);
      alpha[r] = __expf(mrow[r] - mnew);
      float p = __expf(x - mnew);
      float rs = p;
#pragma unroll
      for (int off = 1; off < 16; off <<= 1)
        rs += __shfl_xor(rs, off, 32);
      lrow[r] = lrow[r] * alpha[r] + rs;
      mrow[r] = mnew;
      pshm[(r + 8 * half) * 16 + l16] = p;   // store P in (row, col) order
    }
    __syncthreads();

    // rescale O accumulators by alpha (same row mapping as S)
#pragma unroll
    for (int r = 0; r < 8; ++r) {
      o0[r] *= alpha[r]; o1[r] *= alpha[r];
      o2[r] *= alpha[r]; o3[r] *= alpha[r];
    }

    // ---- O += P * V  (4 K-steps x 4 column chunks) ----
    const float* vblk = v + base + (long)(jb * 16) * HEAD_D;
#pragma unroll
    for (int kp = 0; kp < 4; ++kp) {
      // A slice: P[m=l16][4kp + 2*half + r]
      v2f a = *(const v2f*)&pshm[l16 * 16 + 4 * kp + 2 * half];
      const float* vr = vblk + (long)(4 * kp + 2 * half) * HEAD_D + l16;
      v2f b0, b1, b2, b3;
      b0.x = vr[0];  b0.y = vr[HEAD_D + 0];
      b1.x = vr[16]; b1.y = vr[HEAD_D + 16];
      b2.x = vr[32]; b2.y = vr[HEAD_D + 32];
      b3.x = vr[48]; b3.y = vr[HEAD_D + 48];
      o0 = __builtin_amdgcn_wmma_f32_16x16x4_f32(false, a, false, b0, (short)0, o0, false, false);
      o1 = __builtin_amdgcn_wmma_f32_16x16x4_f32(false, a, false, b1, (short)0, o1, false, false);
      o2 = __builtin_amdgcn_wmma_f32_16x16x4_f32(false, a, false, b2, (short)0, o2, false, false);
      o3 = __builtin_amdgcn_wmma_f32_16x16x4_f32(false, a, false, b3, (short)0, o3, false, false);
    }
    __syncthreads();
  }

  // ---- finalize: divide by row sums and store ----
  float* orow = out + base + (long)(row * 16) * HEAD_D;
#pragma unroll
  for (int r = 0; r < 8; ++r) {
    const float inv = 1.0f / lrow[r];
    const int m = r + 8 * half;
    orow[(long)m * HEAD_D +  0 + l16] = o0[r] * inv;
    orow[(long)m * HEAD_D + 16 + l16] = o1[r] * inv;
    orow[(long)m * HEAD_D + 32 + l16] = o2[r] * inv;
    orow[(long)m * HEAD_D + 48 + l16] = o3[r] * inv;
  }
}

extern "C" void kernel_launch(void* const* d_in, const int* in_sizes, int n_in,
                              void* d_out, int out_size, void* d_ws, size_t ws_size,
                              hipStream_t stream) {
  (void)n_in; (void)out_size; (void)d_ws; (void)ws_size;
  const float* q = (const float*)d_in[0];
  const float* k = (const float*)d_in[1];
  const float* v = (const float*)d_in[2];
  float* out = (float*)d_out;

  const int bh = in_sizes[0] / (SEQ_L * HEAD_D);   // B*H = 32
  dim3 grid((unsigned)(bh * NBLK));                // 4096 single-wave workgroups
  sparse_attn_wmma_f32<<<grid, 32, 0, stream>>>(q, k, v, out);
}